// AnomalyAttention_46514495816143
// MI455X (gfx1250) — compile-verified
//
#include <hip/hip_runtime.h>
#include <math.h>

// Problem dims (fixed by reference)
#define B_ 4
#define L_ 1024
#define H_ 8
#define E_ 64

typedef float v2f __attribute__((ext_vector_type(2)));
typedef float v8f __attribute__((ext_vector_type(8)));

__device__ __forceinline__ v8f wmma_f32_4(v2f a, v2f b, v8f c) {
  // D = A(16x4 f32) * B(4x16 f32) + C(16x16 f32)
  return __builtin_amdgcn_wmma_f32_16x16x4_f32(
      /*neg_a=*/false, a, /*neg_b=*/false, b,
      /*c_mod=*/(short)0, c, /*reuse_a=*/false, /*reuse_b=*/false);
}

// ---------------------------------------------------------------------------
// Kernel 1: prior + sig  (pure streaming writes: 2 x 134 MB)
// one block per (b,h,l) row; 256 threads x float4 = 1024 columns
// ---------------------------------------------------------------------------
__global__ __launch_bounds__(256)
void anomaly_prior_kernel(const float* __restrict__ sigma,
                          float* __restrict__ outPrior,
                          float* __restrict__ outSig) {
  const int row = blockIdx.x;        // bh * L + l
  const int l  = row % L_;
  const int bh = row / L_;
  const int h  = bh % H_;
  const int b  = bh / H_;

  // sig = 3^(sigmoid(5x)+1e-5) - 1
  const float x  = sigma[((size_t)b * L_ + l) * H_ + h];
  const float sg = 1.0f / (1.0f + __expf(-5.0f * x)) + 1e-5f;
  const float sv = __expf(sg * 1.0986122886681098f) - 1.0f;   // 3^sg - 1
  const float invsig = 1.0f / sv;
  const float coef = 0.3989422804014327f * invsig;            // 1/sqrt(2pi)/sig
  const float qq   = -0.5f * invsig * invsig;

  const size_t base = ((size_t)bh * L_ + l) * L_;
  const int s = threadIdx.x * 4;

  float d0 = (float)(l - s),     d1 = (float)(l - (s + 1));
  float d2 = (float)(l - (s + 2)), d3 = (float)(l - (s + 3));
  float4 pr;
  pr.x = coef * __expf(qq * d0 * d0);
  pr.y = coef * __expf(qq * d1 * d1);
  pr.z = coef * __expf(qq * d2 * d2);
  pr.w = coef * __expf(qq * d3 * d3);

  *(float4*)(outPrior + base + s) = pr;
  *(float4*)(outSig   + base + s) = make_float4(sv, sv, sv, sv);
}

// ---------------------------------------------------------------------------
// Kernel 2: causal attention with fp32 WMMA
// one block (8 wave32s) per (b, h, 16-row l-block)
// ---------------------------------------------------------------------------
__global__ __launch_bounds__(256)
void anomaly_attn_kernel(const float* __restrict__ Q,
                         const float* __restrict__ K,
                         const float* __restrict__ Vals,
                         float* __restrict__ outV,
                         float* __restrict__ outSeries) {
  __shared__ float rowbuf[16 * 1024];          // 64 KB score/series strip

  const int nLB  = L_ / 16;
  const int blk_l = blockIdx.x % nLB;
  const int bh    = blockIdx.x / nLB;
  const int h = bh % H_;
  const int b = bh / H_;
  const int l0 = blk_l * 16;

  const int tid  = threadIdx.x;
  const int w    = tid >> 5;        // wave id 0..7
  const int lane = tid & 31;
  const int hi   = lane >> 4;       // 0: lanes 0-15, 1: lanes 16-31
  const int ln   = lane & 15;

  const size_t rowStride = (size_t)H_ * E_;     // 512 floats between l rows
  const float* Qbase = Q    + ((size_t)b * L_) * rowStride + (size_t)h * E_;
  const float* Kbase = K    + ((size_t)b * L_) * rowStride + (size_t)h * E_;
  const float* Vbase = Vals + ((size_t)b * L_) * rowStride + (size_t)h * E_;

  const float NEG_INF = -__builtin_inff();

  // ---- preload Q block as WMMA A-fragments: lane holds row (ln), K-pair (2*hi)
  v2f qa[16];
  {
    const float* qrow = Qbase + (size_t)(l0 + ln) * rowStride + 2 * hi;
#pragma unroll
    for (int k = 0; k < 16; ++k)
      qa[k] = *(const v2f*)(qrow + 4 * k);
  }

  // =======================================================================
  // Phase 1: scores = scale * Q K^T for causal s-tiles, masked, -> LDS
  // =======================================================================
  for (int st = w; st <= blk_l; st += 8) {
    const float* krow = Kbase + (size_t)(st * 16 + ln) * rowStride + 2 * hi;
    if (st + 8 <= blk_l)
      __builtin_prefetch(Kbase + (size_t)((st + 8) * 16 + ln) * rowStride, 0, 3);

    v8f c = {};
#pragma unroll
    for (int k = 0; k < 16; ++k) {
      v2f bv = *(const v2f*)(krow + 4 * k);    // B = K^T fragment
      c = wmma_f32_4(qa[k], bv, c);
    }
    const int diag = (st == blk_l);
#pragma unroll
    for (int v = 0; v < 8; ++v) {
      const int m = v + 8 * hi;                // output row within block
      float val = c[v] * 0.125f;               // 1/sqrt(E)
      if (diag && ln > m) val = NEG_INF;       // causal mask (s > l)
      rowbuf[m * 1024 + st * 16 + ln] = val;
    }
  }
  __syncthreads();

  // =======================================================================
  // Phase 2: row softmax (16 lanes per row), write series (coalesced f4)
  // =======================================================================
  {
    const int r = 2 * w + hi;                  // row 0..15 of the block
    const int ncols = (blk_l + 1) * 16;        // columns actually computed
    const int cbase = ln * 4;
    float* myrow = rowbuf + r * 1024;

    float mx = NEG_INF;
    for (int col = cbase; col < ncols; col += 64) {
      float4 xv = *(float4*)(myrow + col);
      mx = fmaxf(mx, fmaxf(fmaxf(xv.x, xv.y), fmaxf(xv.z, xv.w)));
    }
#pragma unroll
    for (int d = 8; d >= 1; d >>= 1) mx = fmaxf(mx, __shfl_xor(mx, d, 32));

    float sum = 0.0f;
    for (int col = cbase; col < ncols; col += 64) {
      float4 xv = *(float4*)(myrow + col);
      xv.x = __expf(xv.x - mx); xv.y = __expf(xv.y - mx);
      xv.z = __expf(xv.z - mx); xv.w = __expf(xv.w - mx);
      *(float4*)(myrow + col) = xv;
      sum += xv.x + xv.y + xv.z + xv.w;
    }
#pragma unroll
    for (int d = 8; d >= 1; d >>= 1) sum += __shfl_xor(sum, d, 32);
    const float inv = 1.0f / sum;

    float* srow = outSeries + ((size_t)bh * L_ + (l0 + r)) * L_;
    for (int col = cbase; col < L_; col += 64) {
      float4 p;
      if (col < ncols) {
        float4 xv = *(float4*)(myrow + col);
        p.x = xv.x * inv; p.y = xv.y * inv; p.z = xv.z * inv; p.w = xv.w * inv;
        *(float4*)(myrow + col) = p;           // normalized series for phase 3
      } else {
        p = make_float4(0.0f, 0.0f, 0.0f, 0.0f);
      }
      *(float4*)(srow + col) = p;
    }
  }
  __syncthreads();

  // =======================================================================
  // Phase 3: V = series @ values with WMMA; waves split (e-tile, s-parity)
  // =======================================================================
  const int et = w & 3;                        // e-tile 0..3 (16 cols each)
  const int e0 = et * 16;
  v8f acc = {};
  for (int st = (w >> 2); st <= blk_l; st += 2) {
    const int s0 = st * 16;
#pragma unroll
    for (int k = 0; k < 4; ++k) {
      const int sk = s0 + 4 * k + 2 * hi;
      v2f a = *(const v2f*)(rowbuf + ln * 1024 + sk);      // A = series frag
      const float* vrow = Vbase + (size_t)sk * rowStride + e0 + ln;
      v2f bv;
      bv.x = vrow[0];                                       // values[sk  ][e]
      bv.y = vrow[rowStride];                               // values[sk+1][e]
      acc = wmma_f32_4(a, bv, acc);
    }
  }
  __syncthreads();                 // all waves done reading series from LDS

  // cross-wave pair reduction through reused LDS (first 4 KB of rowbuf)
  float* scr = rowbuf;             // layout: scr[e-tile][16][16]
  if (w >= 4) {
#pragma unroll
    for (int v = 0; v < 8; ++v)
      scr[et * 256 + (v + 8 * hi) * 16 + ln] = acc[v];
  }
  __syncthreads();
  if (w < 4) {
    float* vout = outV + ((size_t)b * L_ + l0) * rowStride + (size_t)h * E_ + e0 + ln;
#pragma unroll
    for (int v = 0; v < 8; ++v) {
      const int m = v + 8 * hi;
      vout[(size_t)m * rowStride] = acc[v] + scr[et * 256 + m * 16 + ln];
    }
  }
}

// ---------------------------------------------------------------------------
extern "C" void kernel_launch(void* const* d_in, const int* in_sizes, int n_in,
                              void* d_out, int out_size, void* d_ws, size_t ws_size,
                              hipStream_t stream) {
  (void)in_sizes; (void)n_in; (void)out_size; (void)d_ws; (void)ws_size;
  const float* q  = (const float*)d_in[0];
  const float* k  = (const float*)d_in[1];
  const float* v  = (const float*)d_in[2];
  const float* sg = (const float*)d_in[3];

  float* out = (float*)d_out;
  const size_t vsz = (size_t)B_ * L_ * H_ * E_;   // 2,097,152
  const size_t msz = (size_t)B_ * H_ * L_ * L_;   // 33,554,432
  float* outV      = out;
  float* outSeries = out + vsz;
  float* outPrior  = outSeries + msz;
  float* outSig    = outPrior + msz;

  hipLaunchKernelGGL(anomaly_prior_kernel, dim3(B_ * H_ * L_), dim3(256), 0, stream,
                     sg, outPrior, outSig);
  hipLaunchKernelGGL(anomaly_attn_kernel, dim3(B_ * H_ * (L_ / 16)), dim3(256), 0, stream,
                     q, k, v, outV, outSeries);
}